// SplineInterpolate1D_29051158790787
// MI455X (gfx1250) — compile-verified
//
#include <hip/hip_runtime.h>
#include <math.h>

typedef float v2f __attribute__((ext_vector_type(2)));
typedef float v8f __attribute__((ext_vector_type(8)));

#define NX 512       // data points
#define NE 8192      // eval points
#define MB 514       // number of basis functions (NX + 2)
#define BC 256       // batch * channels
#define SMOOTH 1e-3f

// knot vector t: 4 copies of x0, interior, 4 copies of x_{NX-1}; t(idx) = xdata(clamp(idx-3))
__device__ __forceinline__ float knot(int idx) {
    int c = idx - 3;
    c = c < 0 ? 0 : (c > (NX - 1) ? (NX - 1) : c);
    return -1.0f + 2.0f * (float)c / (float)(NX - 1);
}

// Cox-de Boor: N[0..3] = cubic basis {i..i+3} at x, knot span l = i+3
__device__ __forceinline__ void deboor4(float x, int i, float* N) {
    int l = i + 3;
    float left[4], right[4];
    N[0] = 1.f; N[1] = 0.f; N[2] = 0.f; N[3] = 0.f;
    #pragma unroll
    for (int j = 1; j <= 3; ++j) {
        left[j]  = x - knot(l + 1 - j);
        right[j] = knot(l + j) - x;
        float saved = 0.f;
        #pragma unroll
        for (int r = 0; r < 3; ++r) {
            if (r < j) {
                float temp = N[r] / (right[r + 1] + left[j - r]);
                N[r] = saved + right[r + 1] * temp;
                saved = left[j - r] * temp;
            }
        }
        N[j] = saved;
    }
}

// K1: basis weights + support start for data (512) and eval (8192) points
__global__ void k_basis(float* __restrict__ wD, int* __restrict__ jD,
                        float* __restrict__ wE, int* __restrict__ jE) {
    int idx = blockIdx.x * blockDim.x + threadIdx.x;
    if (idx < NX) {
        int n = idx;
        float x = -1.0f + 2.0f * (float)n / (float)(NX - 1);
        int i = n > (NX - 2) ? (NX - 2) : n;
        float N[4];
        deboor4(x, i, N);
        wD[n * 4 + 0] = N[0]; wD[n * 4 + 1] = N[1];
        wD[n * 4 + 2] = N[2]; wD[n * 4 + 3] = N[3];
        jD[n] = i;
    } else if (idx < NX + NE) {
        int ne = idx - NX;
        float x = -1.0f + 2.0f * (float)ne / (float)(NE - 1);
        float u = (x + 1.0f) * ((float)(NX - 1) * 0.5f);
        int i = (int)u;
        i = i < 0 ? 0 : (i > (NX - 2) ? (NX - 2) : i);
        float N[4];
        deboor4(x, i, N);
        wE[ne * 4 + 0] = N[0]; wE[ne * 4 + 1] = N[1];
        wE[ne * 4 + 2] = N[2]; wE[ne * 4 + 3] = N[3];
        jE[ne] = i;
    }
}

// K2: lower band of BtB + S*I.  Bl[j*4+q] = BtB(j, j-q)
__global__ void k_band(const float* __restrict__ wD, const int* __restrict__ jD,
                       float* __restrict__ Bl) {
    int idx = blockIdx.x * blockDim.x + threadIdx.x;
    if (idx >= MB * 4) return;
    int j = idx >> 2, q = idx & 3;
    if (j - q < 0) { Bl[idx] = 0.f; return; }
    float s = (q == 0) ? SMOOTH : 0.f;
    int n0 = j - 4; if (n0 < 0) n0 = 0;
    int n1 = j + 1; if (n1 > NX - 1) n1 = NX - 1;
    for (int n = n0; n <= n1; ++n) {
        int a = j - jD[n];
        int b = a - q;
        if (a >= 0 && a < 4 && b >= 0 && b < 4)
            s += wD[n * 4 + a] * wD[n * 4 + b];
    }
    Bl[idx] = s;
}

// K3: banded Cholesky (bandwidth 3), register-rolled last-3-rows window
__global__ void k_chol(const float* __restrict__ Bl, float* __restrict__ Lb) {
    if (threadIdx.x != 0 || blockIdx.x != 0) return;
    float Lm1[4] = {1.f, 0.f, 0.f, 0.f};
    float Lm2[4] = {1.f, 0.f, 0.f, 0.f};
    float Lm3[4] = {1.f, 0.f, 0.f, 0.f};
    for (int j = 0; j < MB; ++j) {
        float Lr[4];
        float a3 = Bl[j * 4 + 3], a2 = Bl[j * 4 + 2], a1 = Bl[j * 4 + 1], a0 = Bl[j * 4 + 0];
        Lr[3] = a3 / Lm3[0];                              // L(j, j-3)
        Lr[2] = (a2 - Lr[3] * Lm2[1]) / Lm2[0];           // L(j, j-2)
        Lr[1] = (a1 - Lr[2] * Lm1[1] - Lr[3] * Lm1[2]) / Lm1[0]; // L(j, j-1)
        float d = a0 - Lr[1] * Lr[1] - Lr[2] * Lr[2] - Lr[3] * Lr[3];
        Lr[0] = sqrtf(d);
        Lb[j * 4 + 0] = Lr[0]; Lb[j * 4 + 1] = Lr[1];
        Lb[j * 4 + 2] = Lr[2]; Lb[j * 4 + 3] = Lr[3];
        #pragma unroll
        for (int e = 0; e < 4; ++e) { Lm3[e] = Lm2[e]; Lm2[e] = Lm1[e]; Lm1[e] = Lr[e]; }
    }
}

// K4: Btz transposed: R[j*BC + bc] = sum_n bx[n, j] * Z[bc, n]
__global__ void k_btz(const float* __restrict__ Z, const float* __restrict__ wD,
                      const int* __restrict__ jD, float* __restrict__ R) {
    int j = blockIdx.x;
    int bc = threadIdx.x;
    float s = 0.f;
    int n0 = j - 4; if (n0 < 0) n0 = 0;
    int n1 = j + 1; if (n1 > NX - 1) n1 = NX - 1;
    for (int n = n0; n <= n1; ++n) {
        int a = j - jD[n];
        if (a >= 0 && a < 4) s += wD[n * 4 + a] * Z[bc * NX + n];
    }
    R[j * BC + bc] = s;
}

// K5: forward/backward banded substitution, in place; one thread per channel
__global__ void k_solve(const float* __restrict__ Lb, float* __restrict__ R) {
    int bc = threadIdx.x;
    float y1 = 0.f, y2 = 0.f, y3 = 0.f;
    for (int j = 0; j < MB; ++j) {
        float s = R[j * BC + bc] - Lb[j * 4 + 1] * y1 - Lb[j * 4 + 2] * y2 - Lb[j * 4 + 3] * y3;
        float y = s / Lb[j * 4 + 0];
        R[j * BC + bc] = y;
        y3 = y2; y2 = y1; y1 = y;
    }
    float x1 = 0.f, x2 = 0.f, x3 = 0.f;
    for (int j = MB - 1; j >= 0; --j) {
        float s = R[j * BC + bc];
        if (j + 1 < MB) s -= Lb[(j + 1) * 4 + 1] * x1;
        if (j + 2 < MB) s -= Lb[(j + 2) * 4 + 2] * x2;
        if (j + 3 < MB) s -= Lb[(j + 3) * 4 + 3] * x3;
        float xv = s / Lb[j * 4 + 0];
        R[j * BC + bc] = xv;
        x3 = x2; x2 = x1; x1 = xv;
    }
}

// K6: WMMA evaluation. D(16bc x 16ne) += A(16x4 coef window) x B(4x16 weights), 2 K-steps.
// A lane layout: A[M=lane%16][K=2*(lane/16)+r]; B: B[K=2*(lane/16)+r][N=lane%16];
// D: D[M=r+8*(lane/16)][N=lane%16].
__global__ void __launch_bounds__(256) k_eval(const float* __restrict__ coefT,
                                              const float* __restrict__ wE,
                                              const int* __restrict__ jE,
                                              float* __restrict__ out) {
    int lane = threadIdx.x & 31;
    int wave = threadIdx.x >> 5;                 // 8 waves per block
    int tile = blockIdx.x * 8 + wave;            // 0..8191
    int bc0 = (tile >> 9) << 4;                  // 16 tiles of bc
    int ne0 = (tile & 511) << 4;                 // 512 tiles of ne
    int nn   = lane & 15;
    int half = lane >> 4;
    int ne = ne0 + nn;
    int j0 = jE[ne];
    float4 w4 = ((const float4*)wE)[ne];
    // window base: support start of first eval point in tile (monotone -> minimum)
    int g = __builtin_amdgcn_readfirstlane(jE[ne0]);
    int del = j0 - g;                             // 0 or 1: shift of this column's support
    v8f acc = {};
    #pragma unroll
    for (int s2 = 0; s2 < 2; ++s2) {
        v2f a, b;
        #pragma unroll
        for (int r = 0; r < 2; ++r) {
            int kk = 4 * s2 + 2 * half + r;      // 0..7 within window
            int jj = g + kk;
            int jc = jj > (MB - 1) ? (MB - 1) : jj;
            a[r] = coefT[jc * BC + bc0 + nn];
            // branchless weight pick: d = kk - del selects w4[d] or 0
            int d = kk - del;
            float m0 = (d == 0) ? 1.f : 0.f;
            float m1 = (d == 1) ? 1.f : 0.f;
            float m2 = (d == 2) ? 1.f : 0.f;
            float m3 = (d == 3) ? 1.f : 0.f;
            b[r] = m0 * w4.x + m1 * w4.y + m2 * w4.z + m3 * w4.w;
        }
        acc = __builtin_amdgcn_wmma_f32_16x16x4_f32(
            /*neg_a=*/false, a, /*neg_b=*/false, b,
            /*c_mod=*/(short)0, acc, /*reuse_a=*/false, /*reuse_b=*/false);
    }
    #pragma unroll
    for (int r = 0; r < 8; ++r) {
        out[(bc0 + r + 8 * half) * NE + ne0 + nn] = acc[r];
    }
}

extern "C" void kernel_launch(void* const* d_in, const int* in_sizes, int n_in,
                              void* d_out, int out_size, void* d_ws, size_t ws_size,
                              hipStream_t stream) {
    const float* Z = (const float*)d_in[0];   // (8,32,512) f32
    float* out = (float*)d_out;               // (256, 8192) f32

    // workspace layout (floats/ints, 16B aligned where needed)
    float* wD = (float*)d_ws;                 // 512*4
    int*   jD = (int*)(wD + NX * 4);          // 512
    float* wE = (float*)(jD + NX);            // 8192*4 (16B aligned: 10240B offset)
    int*   jE = (int*)(wE + NE * 4);          // 8192
    float* Bl = (float*)(jE + NE);            // 514*4 band of BtB
    float* Lb = Bl + MB * 4;                  // 514*4 Cholesky band
    float* R  = Lb + MB * 4;                  // 514*256 rhs -> y -> coef (transposed [j][bc])

    k_basis<<<(NX + NE + 255) / 256, 256, 0, stream>>>(wD, jD, wE, jE);
    k_band <<<(MB * 4 + 255) / 256, 256, 0, stream>>>(wD, jD, Bl);
    k_chol <<<1, 32, 0, stream>>>(Bl, Lb);
    k_btz  <<<MB, BC, 0, stream>>>(Z, wD, jD, R);
    k_solve<<<1, BC, 0, stream>>>(Lb, R);
    k_eval <<<(16 * (NE / 16)) / 8, 256, 0, stream>>>(R, wE, jE, out);  // 1024 blocks
}